// nerf_79147657331010
// MI455X (gfx1250) — compile-verified
//
#include <hip/hip_runtime.h>
#include <math.h>

// CDNA5 / gfx1250, wave32. One wave per ray; segmented exclusive product scan
// done in LOG2 space (native v_log_f32 / v_exp_f32, no ln2 conversion):
// per 256-sample tile, D = L_strictlower(16x16) * V(16x16) via four chained
// V_WMMA_F32_16X16X4_F32; chunk totals combined with DPP16 + v_permlanex16.

typedef __attribute__((ext_vector_type(2))) float v2f;
typedef __attribute__((ext_vector_type(8))) float v8f;

#define A2W_EPS 1e-10f

// ---- cross-lane helpers (pure VALU, no ds_bpermute) -----------------------

// DPP16 row_shr:<N> with zero fill (bound_ctrl=1): lane i gets src[i-N] within
// its 16-lane row, 0 if out of row. CTRL: 0x110 | N.
template <int CTRL>
__device__ __forceinline__ float dpp_shr_zero(float x) {
    int i = __builtin_bit_cast(int, x);
    int r = __builtin_amdgcn_update_dpp(0, i, CTRL, 0xF, 0xF, true);
    return __builtin_bit_cast(float, r);
}

// v_permlanex16 with identity selects: lane i reads lane (i ^ 16).
__device__ __forceinline__ float permx16_swap(float x) {
    int i = __builtin_bit_cast(int, x);
    int r = __builtin_amdgcn_permlanex16(i, i, 0x76543210, 0xfedcba98, false, false);
    return __builtin_bit_cast(float, r);
}

__device__ __forceinline__ float readlane_f32(float x, int lane) {
    return __builtin_bit_cast(float,
        __builtin_amdgcn_readlane(__builtin_bit_cast(int, x), lane));
}

// Raw hardware transcendentals. Safe here: log2 inputs are in [1e-10, 1]
// (clamped, normal, positive); exp2 inputs are <= 0 (no overflow possible).
__device__ __forceinline__ float log2_raw(float x) {
    return __builtin_amdgcn_logf(x);             // v_log_f32
}
__device__ __forceinline__ float exp2_raw(float x) {
    return __builtin_amdgcn_exp2f(x);            // v_exp_f32
}

// ---------------------------------------------------------------------------
// Kernel 1: segment offsets. ray_id is sorted, so starts[r] = lower_bound(r).
// Launch N+1 threads; starts[N] = M. Deterministic, no atomics. ray_id (32MB)
// is L2-resident (192MB), so the ~23-probe searches are cheap.
// ---------------------------------------------------------------------------
__global__ void ray_starts_kernel(const int* __restrict__ ray_id, int M, int N,
                                  int* __restrict__ starts) {
    int r = blockIdx.x * blockDim.x + threadIdx.x;
    if (r > N) return;
    int lo = 0, hi = M;
    while (lo < hi) {
        int mid = (lo + hi) >> 1;
        if (ray_id[mid] < r) lo = mid + 1; else hi = mid;
    }
    starts[r] = lo;
}

// ---------------------------------------------------------------------------
// Kernel 2: per-ray scan. 256 threads = 8 wave32 waves, one ray per wave.
// ---------------------------------------------------------------------------
__global__ __launch_bounds__(256)
void alpha2weight_kernel(const float* __restrict__ alpha,
                         const int*   __restrict__ starts,
                         float* __restrict__ weights,
                         float* __restrict__ alphainv_last,
                         int N) {
    const int lane = threadIdx.x & 31;
    const int wave = threadIdx.x >> 5;
    const int r    = blockIdx.x * 8 + wave;
    if (r >= N) return;                          // wave-uniform guard

    int s = starts[r];
    int e = starts[r + 1];
    s = __builtin_amdgcn_readfirstlane(s);       // scalar loop bounds -> EXEC
    e = __builtin_amdgcn_readfirstlane(e);       // stays all-ones at each WMMA

    const int  n      = lane & 15;               // column index (B/D layout)
    const bool hiHalf = lane >= 16;
    const int  r0     = hiHalf ? 2 : 0;          // K rows this lane holds (A/B)
    const int  mbase  = hiHalf ? 8 : 0;          // D rows this lane holds

    // Strictly-lower-triangular 16x16 operator, split into four 16x4 A-blocks.
    // A layout: lane holds M = lane&15, K = 4k+r0 and 4k+r0+1.
    v2f La[4];
#pragma unroll
    for (int k = 0; k < 4; ++k) {
        La[k].x = ((4 * k + r0)     < n) ? 1.0f : 0.0f;
        La[k].y = ((4 * k + r0 + 1) < n) ? 1.0f : 0.0f;
    }

    float carry = 0.0f;                          // wave-uniform log2-product

    for (int p = s; p < e; p += 256) {
        __builtin_prefetch(alpha + p + 256, 0, 1);   // global_prefetch_b8

        // Load log2 values in B-matrix layout: Bv[k] = rows {4k+r0, 4k+r0+1}.
        v2f Bv[4];
#pragma unroll
        for (int k = 0; k < 4; ++k) {
            int g0 = p + n * 16 + 4 * k + r0;
            int g1 = g0 + 1;
            float a0 = (g0 < e) ? alpha[g0] : 0.0f;  // pad: log2(1)=0, neutral
            float a1 = (g1 < e) ? alpha[g1] : 0.0f;
            Bv[k].x = log2_raw(fmaxf(1.0f - a0, A2W_EPS));
            Bv[k].y = log2_raw(fmaxf(1.0f - a1, A2W_EPS));
        }

        // D = sum_k L_k * V_k  (exclusive column prefix sums, 16x16 f32)
        v8f acc = {0.f, 0.f, 0.f, 0.f, 0.f, 0.f, 0.f, 0.f};
        acc = __builtin_amdgcn_wmma_f32_16x16x4_f32(false, La[0], false, Bv[0],
                                                    (short)0, acc, false, false);
        acc = __builtin_amdgcn_wmma_f32_16x16x4_f32(false, La[1], false, Bv[1],
                                                    (short)0, acc, false, false);
        acc = __builtin_amdgcn_wmma_f32_16x16x4_f32(false, La[2], false, Bv[2],
                                                    (short)0, acc, false, false);
        acc = __builtin_amdgcn_wmma_f32_16x16x4_f32(false, La[3], false, Bv[3],
                                                    (short)0, acc, false, false);

        // Chunk totals: t_c = D[15][c] + V[15][c], valid in lanes 16..31
        // (row 15 = acc[7] of the high half; V[15][c] is Bv[3].y there).
        float t  = acc[7] + Bv[3].y;
        float ts = permx16_swap(t);              // low half fetches from high
        float tv = hiHalf ? t : ts;              // every lane: total of chunk n

        // Inclusive Hillis-Steele scan over the 16 chunk totals, DPP row_shr
        // with zero fill; both 16-lane rows hold identical data/results.
        float incl = tv;
        incl += dpp_shr_zero<0x111>(incl);
        incl += dpp_shr_zero<0x112>(incl);
        incl += dpp_shr_zero<0x114>(incl);
        incl += dpp_shr_zero<0x118>(incl);
        float chunk_off  = dpp_shr_zero<0x111>(incl);   // exclusive offsets
        float tile_total = readlane_f32(incl, 15);      // SGPR, wave-uniform

        // Emit weights: T = exp2(carry + chunk_off + intra), w = T * alpha.
        // D layout: acc[v] = row (mbase+v), column n -> sample n*16+mbase+v.
#pragma unroll
        for (int v = 0; v < 8; ++v) {
            int g = p + n * 16 + mbase + v;
            if (g < e) {
                float T = exp2_raw(carry + chunk_off + acc[v]);  // v_exp_f32
                weights[g] = T * alpha[g];
            }
        }
        carry += tile_total;
    }

    if (lane == 0) alphainv_last[r] = exp2_raw(carry);   // exp2(0)=1: empty rays
}

// ---------------------------------------------------------------------------
extern "C" void kernel_launch(void* const* d_in, const int* in_sizes, int n_in,
                              void* d_out, int out_size, void* d_ws, size_t ws_size,
                              hipStream_t stream) {
    const float* alpha  = (const float*)d_in[0];
    const int*   ray_id = (const int*)d_in[1];
    const int M = in_sizes[0];
    const int N = out_size - M;                  // outputs: [weights M | alphainv N]

    int*   starts   = (int*)d_ws;                // (N+1) ints of scratch
    float* weights  = (float*)d_out;
    float* alphainv = (float*)d_out + M;

    ray_starts_kernel<<<(N + 1 + 255) / 256, 256, 0, stream>>>(ray_id, M, N, starts);
    alpha2weight_kernel<<<(N + 7) / 8, 256, 0, stream>>>(alpha, starts, weights,
                                                         alphainv, N);
}